// ScaledDotProductAttentionWithSentenceNorm_54906861912652
// MI455X (gfx1250) — compile-verified
//
#include <hip/hip_runtime.h>
#include <math.h>

typedef float v2f __attribute__((ext_vector_type(2)));
typedef float v4f __attribute__((ext_vector_type(4)));
typedef float v8f __attribute__((ext_vector_type(8)));

#define B_SZ 4
#define H_SZ 8
#define LQ_SZ 512
#define NB_SZ 32
#define NT_SZ 128
#define D_SZ 64
#define QK_SCALE 0.125f  // 1/sqrt(64)

// Single wrapper so the WMMA call site / signature lives in one place.
// Expected lowering: v_wmma_f32_16x16x4_f32 (VOP3P), D = A(16x4) * B(4x16) + C.
__device__ __forceinline__ v8f wmma_f32_16x16x4(v2f a, v2f b, v8f c) {
  return __builtin_amdgcn_wmma_f32_16x16x4_f32(
      /*neg_a=*/false, a, /*neg_b=*/false, b,
      /*c_mod=*/(short)0, c, /*reuse_a=*/false, /*reuse_b=*/false);
}

// One wave (32 lanes) owns one 16-row q-tile. 4 waves per workgroup share the
// same (b,h), so they reuse the cooperatively staged K/V block in LDS.
// LDS: [0,8192) floats = K tile (xor-swizzled), [8192,16384) = V tile.
// After the QK phase the K region is re-used as per-wave W tiles (skewed).
__global__ __launch_bounds__(128) void sdpa_sentence_kernel(
    const float* __restrict__ q, const float* __restrict__ kg,
    const float* __restrict__ vg, const float* __restrict__ attn_s,
    float* __restrict__ out, float* __restrict__ attn_w_out)
{
  __shared__ float lds[16384];  // 64 KB
  float* Ks = lds;
  float* Vs = lds + 8192;

  const int tid  = threadIdx.x;
  const int lane = tid & 31;
  const int wave = tid >> 5;
  const int hi   = lane >> 4;   // half-wave select (0/1)
  const int lo   = lane & 15;
  float* Ws = lds + wave * 2048;  // 16 x 128 skewed weight tile (aliases Ks)

  const int tile = blockIdx.x * 4 + wave;  // 0..1023
  const int b  = tile >> 8;                // / (H * LQ/16)
  const int h  = (tile >> 5) & 7;
  const int qt = tile & 31;
  const int q0 = qt << 4;
  const int bh = b * H_SZ + h;

  // ---- preload Q A-fragments (V_WMMA_F32_16X16X4_F32 layout) ----
  // lane holds A[row][4*kk + 2*hi + {0,1}], row = lane&15
  v2f qf[16];
  {
    const float* qrow = q + ((size_t)bh * LQ_SZ + q0 + lo) * D_SZ;
#pragma unroll
    for (int kk = 0; kk < 16; ++kk) {
      v2f qv = *(const v2f*)(qrow + kk * 4 + 2 * hi);
      qf[kk] = qv * QK_SCALE;
    }
  }

  v8f oacc[4];
#pragma unroll
  for (int c = 0; c < 4; ++c) oacc[c] = (v8f){};

  for (int n = 0; n < NB_SZ; ++n) {
    const float* kblk = kg + (((size_t)b * NB_SZ + n) * H_SZ + h) * (NT_SZ * D_SZ);
    const float* vblk = vg + (((size_t)b * NB_SZ + n) * H_SZ + h) * (NT_SZ * D_SZ);

    __syncthreads();  // previous iteration done with Vs/Ws
    // ---- stage K/V tiles: coalesced global float4, xor-swizzled LDS groups ----
#pragma unroll
    for (int j = 0; j < 16; ++j) {
      int g   = tid + j * 128;   // float4 index within 128x64 tile
      int row = g >> 4;
      int sw  = ((g ^ row) & 15) << 2;
      *(v4f*)(Ks + row * 64 + sw) = *(const v4f*)(kblk + (size_t)g * 4);
      *(v4f*)(Vs + row * 64 + sw) = *(const v4f*)(vblk + (size_t)g * 4);
    }
    // prefetch next block's K/V (global_prefetch_b8), 256B per thread covers 32KB each
    if (n + 1 < NB_SZ) {
      __builtin_prefetch(kblk + (size_t)H_SZ * NT_SZ * D_SZ + tid * 64, 0, 0);
      __builtin_prefetch(vblk + (size_t)H_SZ * NT_SZ * D_SZ + tid * 64, 0, 0);
    }
    __syncthreads();

    // ---- QK^T: S[16][128] as 8 column tiles of 16x16, K-dim = 64 (16 steps) ----
    v8f S[8];
#pragma unroll
    for (int c = 0; c < 8; ++c) {
      v8f acc = (v8f){};
      const int t = c * 16 + lo;          // token column for this lane
      const float* krow = Ks + t * 64;
      const int tsw = t & 15;
#pragma unroll
      for (int kk = 0; kk < 16; ++kk) {
        // B-frag: lane holds K^T[4kk+2hi + {0,1}][t] = k[t][4kk+2hi + {0,1}]
        v2f bf = *(const v2f*)(krow + (((kk ^ tsw) & 15) << 2) + 2 * hi);
        acc = wmma_f32_16x16x4(qf[kk], bf, acc);
      }
      S[c] = acc;
    }
    __syncthreads();  // all waves done reading Ks -> safe to alias as Ws

    // ---- per-row softmax over 128 tokens ----
    // C/D layout: row m = j + 8*hi, col = c*16 + lo; row data lives in one
    // 16-lane half-wave -> butterfly over lane bits 0..3.
    float rmax[8], rsum[8];
#pragma unroll
    for (int j = 0; j < 8; ++j) {
      float m = S[0][j];
#pragma unroll
      for (int c = 1; c < 8; ++c) m = fmaxf(m, S[c][j]);
      m = fmaxf(m, __shfl_xor(m, 1, 32));
      m = fmaxf(m, __shfl_xor(m, 2, 32));
      m = fmaxf(m, __shfl_xor(m, 4, 32));
      m = fmaxf(m, __shfl_xor(m, 8, 32));
      rmax[j] = m;
    }
#pragma unroll
    for (int j = 0; j < 8; ++j) {
      float s = 0.0f;
#pragma unroll
      for (int c = 0; c < 8; ++c) {
        float e = __expf(S[c][j] - rmax[j]);
        S[c][j] = e;
        s += e;
      }
      s += __shfl_xor(s, 1, 32);
      s += __shfl_xor(s, 2, 32);
      s += __shfl_xor(s, 4, 32);
      s += __shfl_xor(s, 8, 32);
      rsum[j] = s;
    }

    // ---- apply sentence reweighting, emit attn_w, stash W into skewed LDS ----
    float sscale[8];
#pragma unroll
    for (int j = 0; j < 8; ++j) {
      int m = j + 8 * hi;
      sscale[j] = attn_s[((size_t)bh * LQ_SZ + q0 + m) * NB_SZ + n] / rsum[j];
    }
    float* aw = attn_w_out + ((size_t)bh * LQ_SZ + q0) * (NB_SZ * NT_SZ) + (size_t)n * NT_SZ;
#pragma unroll
    for (int c = 0; c < 8; ++c) {
#pragma unroll
      for (int j = 0; j < 8; ++j) {
        int m = j + 8 * hi;
        float w = S[c][j] * sscale[j];
        aw[(size_t)m * (NB_SZ * NT_SZ) + c * 16 + lo] = w;
        Ws[m * 128 + ((c * 16 + lo + 4 * m) & 127)] = w;  // additive skew
      }
    }

    // ---- AV: out[16][64] += W[16][128] @ V[128][64], 4 tiles x 32 K-steps ----
#pragma unroll
    for (int c2 = 0; c2 < 4; ++c2) {
      v8f acc = oacc[c2];
      const int d    = c2 * 16 + lo;
      const int grp4 = d >> 2;
      const int doff = d & 3;
      for (int kk = 0; kk < 32; ++kk) {
        const int tb = kk * 4 + 2 * hi;
        // A-frag: W[row=lo][tb + {0,1}] from skewed row
        v2f af = *(const v2f*)(Ws + lo * 128 + ((tb + 4 * lo) & 127));
        // B-frag: V[tb][d], V[tb+1][d] from xor-swizzled groups
        float b0 = Vs[tb * 64 + (((grp4 ^ tb) & 15) << 2) + doff];
        float b1 = Vs[(tb + 1) * 64 + (((grp4 ^ (tb + 1)) & 15) << 2) + doff];
        v2f bf; bf.x = b0; bf.y = b1;
        acc = wmma_f32_16x16x4(af, bf, acc);
      }
      oacc[c2] = acc;
    }
  }

  // ---- write out[b][q0+m][h*64 + d] ----
#pragma unroll
  for (int c2 = 0; c2 < 4; ++c2) {
#pragma unroll
    for (int j = 0; j < 8; ++j) {
      int m = j + 8 * hi;
      int d = c2 * 16 + lo;
      out[((size_t)b * LQ_SZ + q0 + m) * (H_SZ * D_SZ) + h * D_SZ + d] = oacc[c2][j];
    }
  }
}

extern "C" void kernel_launch(void* const* d_in, const int* in_sizes, int n_in,
                              void* d_out, int out_size, void* d_ws, size_t ws_size,
                              hipStream_t stream) {
  (void)in_sizes; (void)n_in; (void)out_size; (void)d_ws; (void)ws_size;
  const float* q  = (const float*)d_in[0];
  const float* k  = (const float*)d_in[1];
  const float* v  = (const float*)d_in[2];
  const float* s  = (const float*)d_in[3];
  float* out    = (float*)d_out;
  float* attn_w = out + (size_t)B_SZ * LQ_SZ * H_SZ * D_SZ;  // outputs concatenated
  dim3 grid(B_SZ * H_SZ * (LQ_SZ / 16) / 4);  // 256 workgroups x 4 waves
  sdpa_sentence_kernel<<<grid, 128, 0, stream>>>(q, k, v, s, out, attn_w);
}